// RelationGCN_16819091931517
// MI455X (gfx1250) — compile-verified
//
#include <hip/hip_runtime.h>
#include <hip/hip_bf16.h>

typedef float v2f __attribute__((ext_vector_type(2)));
typedef float v8f __attribute__((ext_vector_type(8)));

#define RGCN_R 8
#define RGCN_EPS 1e-5f

// ---------------------------------------------------------------------------
// Phase 1: per-(dst,relation) in-degree counts, then invert to 1/cnt.
// ---------------------------------------------------------------------------
__global__ __launch_bounds__(256)
void rgcn_count_kernel(const int* __restrict__ dst, const int* __restrict__ etype,
                       float* __restrict__ cnt, int n_edges)
{
    int e = blockIdx.x * blockDim.x + threadIdx.x;
    if (e >= n_edges) return;
    size_t seg = (size_t)dst[e] * RGCN_R + etype[e];
    __hip_atomic_fetch_add(cnt + seg, 1.0f, __ATOMIC_RELAXED, __HIP_MEMORY_SCOPE_AGENT);
}

__global__ __launch_bounds__(256)
void rgcn_invert_kernel(float* __restrict__ cnt, int n)
{
    int i = blockIdx.x * blockDim.x + threadIdx.x;
    if (i >= n) return;
    float c = cnt[i];
    cnt[i] = (c > 0.0f) ? (1.0f / c) : 0.0f;
}

// ---------------------------------------------------------------------------
// Phase 2: scatter-mean. One wave (32 lanes) per edge; D/32 floats per lane.
// acc[dst*R + r, :] += x[src, :] * inv[dst*R + r]   (=> holds the mean)
// acc fits in the 192MB global L2, so these f32 atomics stay on-chip.
// ---------------------------------------------------------------------------
template<int D>
__global__ __launch_bounds__(256)
void rgcn_scatter_kernel(const float* __restrict__ feat,
                         const int* __restrict__ src, const int* __restrict__ dst,
                         const int* __restrict__ etype,
                         const float* __restrict__ inv,
                         float* __restrict__ acc, int n_edges)
{
    constexpr int VEC = D / 32;
    const int lane = threadIdx.x & 31;
    const int e = (int)((blockIdx.x * (size_t)blockDim.x + threadIdx.x) >> 5);
    if (e >= n_edges) return;

    const int s = src[e];
    const size_t seg = (size_t)dst[e] * RGCN_R + etype[e];
    const float w = inv[seg];
    const float* __restrict__ fp = feat + (size_t)s * D + lane * VEC;
    float* __restrict__ ap = acc + seg * D + lane * VEC;

    float vals[VEC];
    if constexpr (VEC == 2) {
        const float2 f = *reinterpret_cast<const float2*>(fp);
        vals[0] = f.x; vals[1] = f.y;
    } else {
        const float4 f = *reinterpret_cast<const float4*>(fp);
        vals[0] = f.x; vals[1] = f.y; vals[2] = f.z; vals[3] = f.w;
    }
    #pragma unroll
    for (int i = 0; i < VEC; ++i) {
        __hip_atomic_fetch_add(ap + i, vals[i] * w,
                               __ATOMIC_RELAXED, __HIP_MEMORY_SCOPE_AGENT);
    }
}

// ---------------------------------------------------------------------------
// Phase 3: batched transform via WMMA f32 16x16x4, M-blocked x4.
//   out[n, :] = sum_r acc[n*R+r, :DIN] @ W[r]  +  xin[n, :] @ root  + bias
// One wave computes a 64(node) x 16(out-feature) tile. Four named base
// pointers + immediate offsets (no pointer arrays) so the A loads lower to
// pipelined global_load_b64 with constant offsets; each B-fragment feeds 4
// WMMAs (~1.5 vmem per WMMA).
// A-frag (16x4 f32): lanes 0-15 -> K = {k, k+1}; lanes 16-31 -> K = {k+2, k+3}.
// B-frag (4x16 f32): same K split; N = lane & 15.
// C/D (16x16 f32, 8 VGPRs): VGPR v -> row v (lanes 0-15) / row v+8 (lanes 16-31).
// ---------------------------------------------------------------------------
#define RGCN_WMMA(C, A, B) \
    __builtin_amdgcn_wmma_f32_16x16x4_f32(false, (A), false, (B), (short)0, (C), false, false)

template<int DIN, int DOUT>
__global__ __launch_bounds__(256)
void rgcn_gemm_kernel(const float* __restrict__ acc,   // [n*R, DIN] relation means
                      const float* __restrict__ xin,   // [n, DIN]  self features
                      const float* __restrict__ W,     // [R, DIN, DOUT]
                      const float* __restrict__ root,  // [DIN, DOUT]
                      const float* __restrict__ bias,  // [DOUT]
                      float* __restrict__ out,         // [n, DOUT]
                      int n_nodes)
{
    constexpr int NT = DOUT / 16;   // 16-wide column tiles
    constexpr int MB = 4;           // node tiles per wave (M = 64)
    const int lane = threadIdx.x & 31;
    const int wave = (int)((blockIdx.x * (size_t)blockDim.x + threadIdx.x) >> 5);
    const int super_tile = wave / NT;
    const int n_tile     = wave % NT;
    const int node0 = super_tile * (MB * 16);
    if (node0 >= n_nodes) return;   // wave-uniform: EXEC stays all-ones for WMMA

    const int m     = lane & 15;
    const int khalf = lane >> 4;            // 0 -> K offset 0, 1 -> K offset 2
    const int kof   = khalf * 2;
    const int col   = n_tile * 16 + m;

    int n0 = node0 + m;       if (n0 >= n_nodes) n0 = n_nodes - 1;
    int n1 = node0 + 16 + m;  if (n1 >= n_nodes) n1 = n_nodes - 1;
    int n2 = node0 + 32 + m;  if (n2 >= n_nodes) n2 = n_nodes - 1;
    int n3 = node0 + 48 + m;  if (n3 >= n_nodes) n3 = n_nodes - 1;

    // Fixed global base pointers; all further addressing is int offsets.
    const float* __restrict__ a0 = acc + (size_t)n0 * (RGCN_R * DIN);
    const float* __restrict__ a1 = acc + (size_t)n1 * (RGCN_R * DIN);
    const float* __restrict__ a2 = acc + (size_t)n2 * (RGCN_R * DIN);
    const float* __restrict__ a3 = acc + (size_t)n3 * (RGCN_R * DIN);

    v8f c0 = {0.f,0.f,0.f,0.f,0.f,0.f,0.f,0.f};
    v8f c1 = c0, c2 = c0, c3 = c0;

    // --- R relation matrices: A from acc, B from W[mat] ---
    for (int mat = 0; mat < RGCN_R; ++mat) {
        const float* __restrict__ bcol = W + (size_t)mat * (DIN * DOUT) + col;
        const int ab = mat * DIN + kof;
        #pragma unroll
        for (int k = 0; k < DIN; k += 4) {
            v2f b;
            b.x = bcol[(k + kof) * DOUT];
            b.y = bcol[(k + kof + 1) * DOUT];
            const v2f q0 = *reinterpret_cast<const v2f*>(a0 + ab + k);
            const v2f q1 = *reinterpret_cast<const v2f*>(a1 + ab + k);
            const v2f q2 = *reinterpret_cast<const v2f*>(a2 + ab + k);
            const v2f q3 = *reinterpret_cast<const v2f*>(a3 + ab + k);
            c0 = RGCN_WMMA(c0, q0, b);
            c1 = RGCN_WMMA(c1, q1, b);
            c2 = RGCN_WMMA(c2, q2, b);
            c3 = RGCN_WMMA(c3, q3, b);
        }
    }

    // --- root (self) term: A from xin, B from root ---
    {
        const float* __restrict__ r0 = xin + (size_t)n0 * DIN;
        const float* __restrict__ r1 = xin + (size_t)n1 * DIN;
        const float* __restrict__ r2 = xin + (size_t)n2 * DIN;
        const float* __restrict__ r3 = xin + (size_t)n3 * DIN;
        const float* __restrict__ bcol = root + col;
        #pragma unroll
        for (int k = 0; k < DIN; k += 4) {
            v2f b;
            b.x = bcol[(k + kof) * DOUT];
            b.y = bcol[(k + kof + 1) * DOUT];
            const v2f q0 = *reinterpret_cast<const v2f*>(r0 + k + kof);
            const v2f q1 = *reinterpret_cast<const v2f*>(r1 + k + kof);
            const v2f q2 = *reinterpret_cast<const v2f*>(r2 + k + kof);
            const v2f q3 = *reinterpret_cast<const v2f*>(r3 + k + kof);
            c0 = RGCN_WMMA(c0, q0, b);
            c1 = RGCN_WMMA(c1, q1, b);
            c2 = RGCN_WMMA(c2, q2, b);
            c3 = RGCN_WMMA(c3, q3, b);
        }
    }

    const float bv = bias[col];
    #pragma unroll
    for (int v = 0; v < 8; ++v) {
        const int r0w = node0 + v + khalf * 8;
        if (r0w < n_nodes)           out[(size_t)r0w * DOUT + col] = c0[v] + bv;
        const int r1w = r0w + 16;
        if (r1w < n_nodes)           out[(size_t)r1w * DOUT + col] = c1[v] + bv;
        const int r2w = r0w + 32;
        if (r2w < n_nodes)           out[(size_t)r2w * DOUT + col] = c2[v] + bv;
        const int r3w = r0w + 48;
        if (r3w < n_nodes)           out[(size_t)r3w * DOUT + col] = c3[v] + bv;
    }
}

// ---------------------------------------------------------------------------
// Phase 4: optional LeakyReLU(0.2) + LayerNorm. One wave per node (wave32).
// ---------------------------------------------------------------------------
template<int D, bool LRELU>
__global__ __launch_bounds__(256)
void rgcn_ln_kernel(const float* __restrict__ in, float* __restrict__ out,
                    const float* __restrict__ gamma, const float* __restrict__ beta,
                    int n_nodes)
{
    constexpr int VEC = D / 32;
    const int lane = threadIdx.x & 31;
    const int node = (int)((blockIdx.x * (size_t)blockDim.x + threadIdx.x) >> 5);
    if (node >= n_nodes) return;

    float v[VEC];
    const float* __restrict__ ip = in + (size_t)node * D;
    float s = 0.f, sq = 0.f;
    #pragma unroll
    for (int i = 0; i < VEC; ++i) {
        float t = ip[lane * VEC + i];
        if (LRELU) t = (t >= 0.f) ? t : 0.2f * t;
        v[i] = t; s += t; sq += t * t;
    }
    #pragma unroll
    for (int off = 16; off > 0; off >>= 1) {
        s  += __shfl_xor(s,  off, 32);
        sq += __shfl_xor(sq, off, 32);
    }
    const float mu   = s / (float)D;
    const float var  = sq / (float)D - mu * mu;
    const float rstd = rsqrtf(var + RGCN_EPS);
    float* __restrict__ op = out + (size_t)node * D;
    #pragma unroll
    for (int i = 0; i < VEC; ++i) {
        const int j = lane * VEC + i;
        op[j] = (v[i] - mu) * rstd * gamma[j] + beta[j];
    }
}

// ---------------------------------------------------------------------------
// Host-side orchestration (graph-capture safe: only kernel launches and
// hipMemsetAsync on `stream`; fully deterministic per call).
// ---------------------------------------------------------------------------
extern "C" void kernel_launch(void* const* d_in, const int* in_sizes, int n_in,
                              void* d_out, int out_size, void* d_ws, size_t ws_size,
                              hipStream_t stream) {
    (void)n_in; (void)out_size; (void)ws_size;

    const float* x      = (const float*)d_in[0];
    const int*   eidx   = (const int*)  d_in[1];
    const int*   etype  = (const int*)  d_in[2];
    const float* W1     = (const float*)d_in[3];
    const float* root1  = (const float*)d_in[4];
    const float* bias1  = (const float*)d_in[5];
    const float* gamma1 = (const float*)d_in[6];
    const float* beta1  = (const float*)d_in[7];
    const float* W2     = (const float*)d_in[8];
    const float* root2  = (const float*)d_in[9];
    const float* bias2  = (const float*)d_in[10];
    const float* gamma2 = (const float*)d_in[11];
    const float* beta2  = (const float*)d_in[12];
    float* out = (float*)d_out;

    const int E = in_sizes[2];
    const int n = in_sizes[0] / 64;            // D_IN = 64
    const int* src = eidx;
    const int* dst = eidx + E;

    const size_t NR = (size_t)n * RGCN_R;

    // Workspace layout (256B aligned regions):
    //   cnt/inv : NR floats                (1.6 MB)
    //   h       : n*128 floats             (25.6 MB)  layer-1 activations
    //   acc     : NR*128 floats            (204.8 MB) reused for both layers
    char* ws = (char*)d_ws;
    auto align256 = [](size_t v) { return (v + 255) & ~(size_t)255; };
    float* cnt = (float*)ws;
    size_t off = align256(NR * sizeof(float));
    float* h = (float*)(ws + off);
    off += align256((size_t)n * 128 * sizeof(float));
    float* acc = (float*)(ws + off);

    const int T = 256;
    const int edge_blocks      = (E + T - 1) / T;
    const int seg_blocks       = (int)((NR + T - 1) / T);
    const int edge_wave_blocks = (E + 7) / 8;          // 8 waves/block, wave/edge
    const int super_tiles      = (n + 63) / 64;        // 64 nodes per wave (MB=4)
    const int ln_blocks        = (n + 7) / 8;          // 8 waves/block, wave/node

    // --- shared in-degree normalization ---
    hipMemsetAsync(cnt, 0, NR * sizeof(float), stream);
    rgcn_count_kernel<<<edge_blocks, T, 0, stream>>>(dst, etype, cnt, E);
    rgcn_invert_kernel<<<seg_blocks, T, 0, stream>>>(cnt, (int)NR);

    // --- layer 1: aggregate (D=64) -> WMMA transform (64->128) -> LReLU+LN ---
    hipMemsetAsync(acc, 0, NR * 64 * sizeof(float), stream);
    rgcn_scatter_kernel<64><<<edge_wave_blocks, T, 0, stream>>>(
        x, src, dst, etype, cnt, acc, E);
    {
        const int waves  = super_tiles * (128 / 16);
        const int blocks = (waves + 7) / 8;
        rgcn_gemm_kernel<64, 128><<<blocks, T, 0, stream>>>(
            acc, x, W1, root1, bias1, h, n);
    }
    rgcn_ln_kernel<128, true><<<ln_blocks, T, 0, stream>>>(h, h, gamma1, beta1, n);

    // --- layer 2: aggregate (D=128) -> WMMA transform (128->64) -> LN ---
    hipMemsetAsync(acc, 0, NR * 128 * sizeof(float), stream);
    rgcn_scatter_kernel<128><<<edge_wave_blocks, T, 0, stream>>>(
        h, src, dst, etype, cnt, acc, E);
    {
        const int waves  = super_tiles * (64 / 16);
        const int blocks = (waves + 7) / 8;
        rgcn_gemm_kernel<128, 64><<<blocks, T, 0, stream>>>(
            acc, h, W2, root2, bias2, out, n);
    }
    rgcn_ln_kernel<64, false><<<ln_blocks, T, 0, stream>>>(out, out, gamma2, beta2, n);
}